// MyGCNlayer_2568390443564
// MI455X (gfx1250) — compile-verified
//
#include <hip/hip_runtime.h>

typedef __attribute__((ext_vector_type(2))) float v2f;
typedef __attribute__((ext_vector_type(8))) float v8f;

#define GCN_D 64

// ---------------------------------------------------------------------------
// y = x @ W^T via V_WMMA_F32_16X16X4_F32, one wave32 per 16x16 output tile.
//
// A-fragment (16x4 f32, 2 VGPRs/lane) per CDNA5 ISA 7.12.2:
//   lanes 0-15 : M = lane,    {a.x,a.y} = K = {0,1}  (within the 4-wide block)
//   lanes 16-31: M = lane-16, {a.x,a.y} = K = {2,3}
// B-fragment (4x16 f32, 2 VGPRs/lane), mirrored with lane%16 = N.
// C/D (16x16 f32, 8 VGPRs): VGPR i -> M = i + 8*(lane>=16), N = lane%16.
// ---------------------------------------------------------------------------
__global__ __launch_bounds__(256) void gcn_gemm_wmma(
    const float* __restrict__ x, const float* __restrict__ W,
    float* __restrict__ y, int n_nodes) {
  const int lane  = threadIdx.x & 31;
  const int gwave = (blockIdx.x * blockDim.x + threadIdx.x) >> 5;
  const int tiles_n = GCN_D / 16;                 // 4
  const int m0 = (gwave / tiles_n) * 16;
  const int n0 = (gwave % tiles_n) * 16;
  if (m0 >= n_nodes) return;

  const int lhalf = lane & 15;
  const int khalf = (lane >> 4) << 1;             // 0 or 2
  int arow = m0 + lhalf;
  if (arow >= n_nodes) arow = n_nodes - 1;        // clamp (tail tiles only)
  const float* __restrict__ xrow = x + (size_t)arow * GCN_D;
  const int ncol = n0 + lhalf;                    // output column for this lane

  v8f c = {};
#pragma unroll
  for (int kb = 0; kb < GCN_D / 4; ++kb) {
    const int k = (kb << 2) + khalf;
    v2f a;
    a.x = xrow[k];
    a.y = xrow[k + 1];
    // B = W^T : B[k][n] = W[n][k] = W[n*D + k]
    v2f b;
    b.x = W[ncol * GCN_D + k];
    b.y = W[ncol * GCN_D + k + 1];
    c = __builtin_amdgcn_wmma_f32_16x16x4_f32(
        /*neg_a=*/false, a, /*neg_b=*/false, b,
        /*c_mod=*/(short)0, c, /*reuse_a=*/false, /*reuse_b=*/false);
  }

  const int row_base = m0 + ((lane >> 4) << 3);   // +8 for upper half-lanes
  float* __restrict__ yp = y + (size_t)row_base * GCN_D + ncol;
  if (m0 + 16 <= n_nodes) {
    // full tile: unconditional strided stores, no per-row exec masking
#pragma unroll
    for (int i = 0; i < 8; ++i) yp[(size_t)i * GCN_D] = c[i];
  } else {
#pragma unroll
    for (int i = 0; i < 8; ++i) {
      if (row_base + i < n_nodes) yp[(size_t)i * GCN_D] = c[i];
    }
  }
}

// out[n][c] = bias[c], vectorized: one float4 (global_store_b128) per thread.
__global__ __launch_bounds__(256) void gcn_init(
    float4* __restrict__ out4, const float* __restrict__ bias, int total4) {
  const int i = blockIdx.x * blockDim.x + threadIdx.x;
  if (i < total4) {
    const float4 b = *(const float4*)(bias + ((i & (GCN_D / 4 - 1)) << 2));
    out4[i] = b;
  }
}

// 16 threads per edge, each covering 4 consecutive columns:
//   float4 gather of y[src]  -> global_load_b128 (wave covers 2 edges, 512 B)
//   4x global_atomic_add_f32 into out[dst]
// y and out are both L2-resident (25.6 MB each, 192 MB L2), so this phase is
// L2-bandwidth bound; only the edge-index stream comes from HBM.
__global__ __launch_bounds__(256) void gcn_scatter(
    const float* __restrict__ y, const int* __restrict__ src,
    const int* __restrict__ dst, float* __restrict__ out, int n_edges) {
  const long long t = (long long)blockIdx.x * blockDim.x + threadIdx.x;
  const long long total = (long long)n_edges * 16;
  if (t >= total) return;
  const int e    = (int)(t >> 4);
  const int c0   = (int)(t & 15) << 2;            // column base (16B aligned)
  const int s = src[e];
  const int d = dst[e];
  const float4 v = *(const float4*)(y + (size_t)s * GCN_D + c0);
  float* o = out + (size_t)d * GCN_D + c0;
  unsafeAtomicAdd(o + 0, v.x);
  unsafeAtomicAdd(o + 1, v.y);
  unsafeAtomicAdd(o + 2, v.z);
  unsafeAtomicAdd(o + 3, v.w);
}

extern "C" void kernel_launch(void* const* d_in, const int* in_sizes, int n_in,
                              void* d_out, int out_size, void* d_ws, size_t ws_size,
                              hipStream_t stream) {
  const float* x    = (const float*)d_in[0];   // [N, 64]
  const float* W    = (const float*)d_in[1];   // [64, 64]
  const float* bias = (const float*)d_in[2];   // [64]
  const int* eidx   = (const int*)d_in[3];     // [2, E] flat

  const int n_nodes = in_sizes[0] / GCN_D;
  const int n_edges = in_sizes[3] / 2;
  const int* src = eidx;
  const int* dst = eidx + n_edges;

  float* out = (float*)d_out;
  float* y   = (float*)d_ws;                   // N*64 f32 scratch (25.6 MB)

  // 1) GEMM: one wave per 16x16 tile, 8 waves per 256-thread block
  const int tiles = ((n_nodes + 15) / 16) * (GCN_D / 16);
  const int gemm_blocks = (tiles + 7) / 8;
  gcn_gemm_wmma<<<gemm_blocks, 256, 0, stream>>>(x, W, y, n_nodes);

  // 2) out = bias (zeros in the reference, kept general)
  const int total4 = n_nodes * (GCN_D / 4);
  gcn_init<<<(total4 + 255) / 256, 256, 0, stream>>>((float4*)out, bias, total4);

  // 3) edge gather + atomic scatter-sum
  const long long work = (long long)n_edges * 16;
  const int sblocks = (int)((work + 255) / 256);
  gcn_scatter<<<sblocks, 256, 0, stream>>>(y, src, dst, out, n_edges);
}